// GCNEncoder_Commoninput_35785667510356
// MI455X (gfx1250) — compile-verified
//
#include <hip/hip_runtime.h>
#include <math.h>

// ---------------------------------------------------------------------------
// Types for CDNA5 WMMA (fp32 16x16x4)
// ---------------------------------------------------------------------------
typedef float v2f __attribute__((ext_vector_type(2)));
typedef float v8f __attribute__((ext_vector_type(8)));

// ---------------------------------------------------------------------------
// out[M x DOUT] = x[M x DIN] @ W^T + bias,  W is [DOUT x DIN] row-major.
// One wave computes a 16 x DOUT strip using V_WMMA_F32_16X16X4_F32.
// A-fragment (16x4 f32): lanes 0-15 rows M=0..15 with K={k0,k0+1},
//                        lanes 16-31 same rows with K={k0+2,k0+3}.
// B-fragment (4x16 f32): mirrored; C/D per ISA 8-VGPR layout.
// ---------------------------------------------------------------------------
template <int DIN, int DOUT>
__global__ void proj_wmma(const float* __restrict__ x,
                          const float* __restrict__ Wm,
                          const float* __restrict__ bias,
                          float* __restrict__ out, int nrows) {
  constexpr int NT = DOUT / 16;
  const int wave  = threadIdx.x >> 5;
  const int lane  = threadIdx.x & 31;
  const int mTile = blockIdx.x * (blockDim.x >> 5) + wave;
  const int mTiles = nrows >> 4;
  if (mTile >= mTiles) return;

  const int rlo = lane & 15;   // row within tile (A) / col within tile (B,D)
  const int kh  = lane >> 4;   // K-half select (0 -> K=0,1 ; 1 -> K=2,3)

  v8f acc[NT];
  v8f zero = {};
  #pragma unroll
  for (int t = 0; t < NT; ++t) acc[t] = zero;

  const float* xrow = x + (size_t)((mTile << 4) + rlo) * DIN + 2 * kh;

  for (int k0 = 0; k0 < DIN; k0 += 4) {
    v2f a;
    a.x = xrow[k0];
    a.y = xrow[k0 + 1];
    #pragma unroll
    for (int t = 0; t < NT; ++t) {
      const float* wrow = Wm + (size_t)(t * 16 + rlo) * DIN + k0 + 2 * kh;
      v2f b;
      b.x = wrow[0];
      b.y = wrow[1];
      acc[t] = __builtin_amdgcn_wmma_f32_16x16x4_f32(
          /*neg_a=*/false, a, /*neg_b=*/false, b,
          /*c_mod=*/(short)0, acc[t], /*reuse_a=*/false, /*reuse_b=*/false);
    }
  }

  // D layout: VGPR r -> lanes 0-15: (M=r, N=lane); lanes 16-31: (M=r+8, N=lane-16)
  #pragma unroll
  for (int t = 0; t < NT; ++t) {
    #pragma unroll
    for (int r = 0; r < 8; ++r) {
      const int m = (mTile << 4) + r + 8 * kh;
      const int n = t * 16 + rlo;
      out[(size_t)m * DOUT + n] = acc[t][r] + bias[n];
    }
  }
}

// ---------------------------------------------------------------------------
// Float atomic max via integer punning (init value must be -inf)
// ---------------------------------------------------------------------------
__device__ __forceinline__ void atomicMaxFloat(float* addr, float v) {
  if (v >= 0.0f)
    atomicMax((int*)addr, __float_as_int(v));
  else
    atomicMin((unsigned int*)addr, __float_as_uint(v));
}

// ---------------------------------------------------------------------------
// Edge pass 1: logits[e,h] = dot(q[dst,h,:], k[src,h,:]) / sqrt(32)
// One wave32 per edge; lane == channel within each 32-wide head.
// ---------------------------------------------------------------------------
template <int H>
__global__ void edge_logits(const float* __restrict__ q,
                            const float* __restrict__ k,
                            const int* __restrict__ src,
                            const int* __restrict__ dst,
                            float* __restrict__ logit,
                            float* __restrict__ nodemax, int nedges) {
  constexpr int F = H * 32;
  const int e    = (blockIdx.x * blockDim.x + threadIdx.x) >> 5;
  const int lane = threadIdx.x & 31;
  if (e >= nedges) return;
  const int s = src[e], d = dst[e];
  const float scale = 0.17677669529663687f;  // 1/sqrt(32)
  #pragma unroll
  for (int h = 0; h < H; ++h) {
    float p = q[(size_t)d * F + h * 32 + lane] * k[(size_t)s * F + h * 32 + lane];
    #pragma unroll
    for (int m = 16; m > 0; m >>= 1) p += __shfl_xor(p, m);
    if (lane == 0) {
      const float lg = p * scale;
      logit[(size_t)e * H + h] = lg;
      atomicMaxFloat(&nodemax[(size_t)d * H + h], lg);
    }
  }
}

// ---------------------------------------------------------------------------
// Edge pass 2: unnormalized softmax accumulation.
//   w = exp(logit - max[dst]);  denom[dst] += w;  agg[dst] += w * v[src]
// ---------------------------------------------------------------------------
template <int H>
__global__ void edge_accum(const float* __restrict__ v,
                           const int* __restrict__ src,
                           const int* __restrict__ dst,
                           const float* __restrict__ logit,
                           const float* __restrict__ nodemax,
                           float* __restrict__ denom,
                           float* __restrict__ agg, int nedges) {
  constexpr int F = H * 32;
  const int e    = (blockIdx.x * blockDim.x + threadIdx.x) >> 5;
  const int lane = threadIdx.x & 31;
  if (e >= nedges) return;
  const int s = src[e], d = dst[e];
  #pragma unroll
  for (int h = 0; h < H; ++h) {
    const float w =
        __expf(logit[(size_t)e * H + h] - nodemax[(size_t)d * H + h]);
    if (lane == 0) atomicAdd(&denom[(size_t)d * H + h], w);
    atomicAdd(&agg[(size_t)d * F + h * 32 + lane],
              w * v[(size_t)s * F + h * 32 + lane]);
  }
}

// ---------------------------------------------------------------------------
// Per-node finalize: out = agg / (denom + eps) + skip
// ---------------------------------------------------------------------------
template <int H>
__global__ void finalize_node(const float* __restrict__ skip,
                              const float* __restrict__ denom,
                              const float* __restrict__ agg,
                              float* __restrict__ out, int n) {
  constexpr int F = H * 32;
  const int total = n * F;
  for (int i = blockIdx.x * blockDim.x + threadIdx.x; i < total;
       i += gridDim.x * blockDim.x) {
    const int node = i / F;
    const int c = i - node * F;
    const int h = c >> 5;
    out[i] = agg[i] / (denom[(size_t)node * H + h] + 1e-16f) + skip[i];
  }
}

// ---------------------------------------------------------------------------
// GraphNorm stats over 64 channels: stats[c]=sum x, stats[64+c]=sum x^2
// ---------------------------------------------------------------------------
__global__ void stats64(const float* __restrict__ h, float* __restrict__ stats,
                        int n) {
  __shared__ float s1[64];
  __shared__ float s2[64];
  const int t = threadIdx.x;
  if (t < 64) { s1[t] = 0.0f; s2[t] = 0.0f; }
  __syncthreads();
  const int c = t & 63;
  const int r0 = (blockIdx.x * blockDim.x + t) >> 6;
  const int rs = (gridDim.x * blockDim.x) >> 6;
  float a = 0.0f, b = 0.0f;
  for (int r = r0; r < n; r += rs) {
    const float x = h[(size_t)r * 64 + c];
    a += x;
    b += x * x;
  }
  atomicAdd(&s1[c], a);
  atomicAdd(&s2[c], b);
  __syncthreads();
  if (t < 64) {
    atomicAdd(&stats[t], s1[t]);
    atomicAdd(&stats[64 + t], s2[t]);
  }
}

// ---------------------------------------------------------------------------
// GraphNorm + Mish:
//   xc = x - ms*mean; var = E[x^2] - 2*ms*mean*E[x] + (ms*mean)^2
//   y  = xn * tanh(softplus(xn)),  xn = xc*rsqrt(var+eps)*w + b
// ---------------------------------------------------------------------------
__global__ void norm_mish(const float* __restrict__ h,
                          const float* __restrict__ stats,
                          const float* __restrict__ w,
                          const float* __restrict__ b,
                          const float* __restrict__ ms,
                          float* __restrict__ out, int n) {
  const int total = n * 64;
  const float invN = 1.0f / (float)n;
  for (int i = blockIdx.x * blockDim.x + threadIdx.x; i < total;
       i += gridDim.x * blockDim.x) {
    const int c = i & 63;
    const float mean = stats[c] * invN;
    const float ex2 = stats[64 + c] * invN;
    const float m = ms[c];
    const float var = ex2 - 2.0f * m * mean * mean + m * m * mean * mean;
    const float xc = h[i] - m * mean;
    const float xn = xc * rsqrtf(var + 1e-5f) * w[c] + b[c];
    const float sp = (xn > 20.0f) ? xn : log1pf(__expf(xn));
    out[i] = xn * tanhf(sp);
  }
}

__global__ void fill_f32(float* __restrict__ p, float v, size_t count) {
  size_t i = (size_t)blockIdx.x * blockDim.x + threadIdx.x;
  const size_t stride = (size_t)gridDim.x * blockDim.x;
  for (; i < count; i += stride) p[i] = v;
}

// ---------------------------------------------------------------------------
extern "C" void kernel_launch(void* const* d_in, const int* in_sizes, int n_in,
                              void* d_out, int out_size, void* d_ws,
                              size_t ws_size, hipStream_t stream) {
  const float* x    = (const float*)d_in[0];
  const int*   ei   = (const int*)d_in[1];   // JAX default: int64 -> int32
  const float* Wq1  = (const float*)d_in[2];
  const float* bq1  = (const float*)d_in[3];
  const float* Wk1  = (const float*)d_in[4];
  const float* bk1  = (const float*)d_in[5];
  const float* Wv1  = (const float*)d_in[6];
  const float* bv1  = (const float*)d_in[7];
  const float* Ws1  = (const float*)d_in[8];
  const float* bs1  = (const float*)d_in[9];
  const float* gnw  = (const float*)d_in[10];
  const float* gnb  = (const float*)d_in[11];
  const float* gnms = (const float*)d_in[12];
  const float* Wq2  = (const float*)d_in[13];
  const float* bq2  = (const float*)d_in[14];
  const float* Wk2  = (const float*)d_in[15];
  const float* bk2  = (const float*)d_in[16];
  const float* Wv2  = (const float*)d_in[17];
  const float* bv2  = (const float*)d_in[18];
  const float* Ws2  = (const float*)d_in[19];
  const float* bs2  = (const float*)d_in[20];
  float* out = (float*)d_out;

  const int N = in_sizes[0] / 128;
  const int E = in_sizes[1] / 2;
  const int* srcIdx = ei;       // edge_index[0]
  const int* dstIdx = ei + E;   // edge_index[1]

  // Workspace layout (floats)
  float* ws = (float*)d_ws;
  size_t o = 0;
  float* q   = ws + o; o += (size_t)N * 64;
  float* kk  = ws + o; o += (size_t)N * 64;
  float* vv  = ws + o; o += (size_t)N * 64;
  float* sk  = ws + o; o += (size_t)N * 64;
  float* agg = ws + o; o += (size_t)N * 64;
  float* hn  = ws + o; o += (size_t)N * 64;
  float* lg  = ws + o; o += (size_t)E * 2;
  float* nmx = ws + o; o += (size_t)N * 2;
  float* den = ws + o; o += (size_t)N * 2;
  float* st  = ws + o; o += 128;

  const int projBlk1 = (N / 16 + 7) / 8;   // 8 waves per 256-thread block
  const int edgeBlk  = (E + 7) / 8;        // 8 edges (waves) per block
  const int ewGrid   = 1024;

  // ---------------- Layer 1 (H=2, DIN=128, DOUT=64) ----------------
  proj_wmma<128, 64><<<projBlk1, 256, 0, stream>>>(x, Wq1, bq1, q, N);
  proj_wmma<128, 64><<<projBlk1, 256, 0, stream>>>(x, Wk1, bk1, kk, N);
  proj_wmma<128, 64><<<projBlk1, 256, 0, stream>>>(x, Wv1, bv1, vv, N);
  proj_wmma<128, 64><<<projBlk1, 256, 0, stream>>>(x, Ws1, bs1, sk, N);

  fill_f32<<<512, 256, 0, stream>>>(agg, 0.0f, (size_t)N * 64);
  fill_f32<<<64, 256, 0, stream>>>(den, 0.0f, (size_t)N * 2);
  fill_f32<<<64, 256, 0, stream>>>(nmx, -INFINITY, (size_t)N * 2);
  fill_f32<<<1, 128, 0, stream>>>(st, 0.0f, 128);

  edge_logits<2><<<edgeBlk, 256, 0, stream>>>(q, kk, srcIdx, dstIdx, lg, nmx, E);
  edge_accum<2><<<edgeBlk, 256, 0, stream>>>(vv, srcIdx, dstIdx, lg, nmx, den, agg, E);
  finalize_node<2><<<ewGrid, 256, 0, stream>>>(sk, den, agg, agg, N);

  stats64<<<256, 256, 0, stream>>>(agg, st, N);
  norm_mish<<<ewGrid, 256, 0, stream>>>(agg, st, gnw, gnb, gnms, hn, N);

  // ---------------- Layer 2 (H=1, DIN=64, DOUT=32) ----------------
  proj_wmma<64, 32><<<projBlk1, 256, 0, stream>>>(hn, Wq2, bq2, q, N);
  proj_wmma<64, 32><<<projBlk1, 256, 0, stream>>>(hn, Wk2, bk2, kk, N);
  proj_wmma<64, 32><<<projBlk1, 256, 0, stream>>>(hn, Wv2, bv2, vv, N);
  proj_wmma<64, 32><<<projBlk1, 256, 0, stream>>>(hn, Ws2, bs2, sk, N);

  fill_f32<<<512, 256, 0, stream>>>(agg, 0.0f, (size_t)N * 32);
  fill_f32<<<64, 256, 0, stream>>>(den, 0.0f, (size_t)N);
  fill_f32<<<64, 256, 0, stream>>>(nmx, -INFINITY, (size_t)N);

  edge_logits<1><<<edgeBlk, 256, 0, stream>>>(q, kk, srcIdx, dstIdx, lg, nmx, E);
  edge_accum<1><<<edgeBlk, 256, 0, stream>>>(vv, srcIdx, dstIdx, lg, nmx, den, agg, E);
  finalize_node<1><<<ewGrid, 256, 0, stream>>>(sk, den, agg, out, N);
}